// GCN_58875411693936
// MI455X (gfx1250) — compile-verified
//
#include <hip/hip_runtime.h>

typedef float v2f __attribute__((ext_vector_type(2)));
typedef float v8f __attribute__((ext_vector_type(8)));

#define N_NODES 50000
#define HIDC 64
#define KMAX 128
#define APAD 4   // LDS row pad (floats): lane l reads bank 4*l(+1), halves offset by 2 -> conflict-free

// ---------------- degree / normalization ----------------

__global__ void k_init_deg(float* __restrict__ deg, int n) {
    int i = blockIdx.x * blockDim.x + threadIdx.x;
    if (i < n) deg[i] = 1.0f;              // self-loop contributes 1
}

__global__ void k_count_deg(const int* __restrict__ dst, float* __restrict__ deg, int E) {
    int e = blockIdx.x * blockDim.x + threadIdx.x;
    if (e < E) atomicAdd(&deg[dst[e]], 1.0f);
}

__global__ void k_rsqrt_inplace(float* __restrict__ d, int n) {
    int i = blockIdx.x * blockDim.x + threadIdx.x;
    if (i < n) d[i] = rsqrtf(d[i]);        // deg >= 1 always
}

// ---------------- WMMA f32 GEMM: C[M x N] = A[M x K] @ B[K x N] (+bias) ----
// grid.x = M/16 row tiles; block = 32 * ceil(N/16) threads; wave w -> col tile w.
// A tile (16 x K) staged once per block in padded LDS, shared by all waves.
// Fragment layouts per CDNA5 ISA 7.12.2 (V_WMMA_F32_16X16X4_F32):
//   A 16x4 : lanes 0-15 hold M=lane  K={0,1}; lanes 16-31 hold M=lane-16 K={2,3}
//   B 4x16 : lanes 0-15 hold N=lane  K={0,1}; lanes 16-31 hold N=lane-16 K={2,3}
//   C 16x16: vgpr v, lanes 0-15 -> (M=v,   N=lane); lanes 16-31 -> (M=v+8, N=lane-16)
// B columns >= N are CLAMPED (not masked): they only pollute D columns that
// are never stored, so the K-loop stays branch-free with full EXEC.

__global__ void k_wmma_gemm(const float* __restrict__ A, const float* __restrict__ B,
                            const float* __restrict__ bias, float* __restrict__ C,
                            int K, int N) {
    __shared__ float As[16 * (KMAX + APAD)];

    const int m0   = blockIdx.x * 16;
    const int lane = threadIdx.x & 31;
    const int wave = threadIdx.x >> 5;
    const int n0   = wave * 16;
    const int half = lane >> 4;       // 0: K 0/1,  1: K 2/3
    const int idx  = lane & 15;       // M row (for A) / N col (for B, C)
    const int ldsK = K + APAD;

    // cooperative stage of A tile: 16 rows of K floats (rows contiguous in A)
    {
        const int kq = K >> 2;        // float4s per row (16 or 32)
        for (int r = 0; r < 16; ++r) {
            const float4* srow = (const float4*)(A + (long)(m0 + r) * K);
            float4*       drow = (float4*)(As + r * ldsK);
            for (int c = threadIdx.x; c < kq; c += blockDim.x) drow[c] = srow[c];
        }
    }
    __syncthreads();

    const float* arow = As + idx * ldsK;
    const int    bcol = n0 + idx;
    const bool   bok  = bcol < N;
    const float* bp   = B + (bok ? bcol : (N - 1));   // clamped, always in-bounds

    v8f acc = {0.f, 0.f, 0.f, 0.f, 0.f, 0.f, 0.f, 0.f};

    for (int k = 0; k < K; k += 4) {
        const int ka = k + 2 * half;
        v2f a, b;
        a.x = arow[ka];
        a.y = arow[ka + 1];
        b.x = bp[(long)ka * N];
        b.y = bp[(long)(ka + 1) * N];
        acc = __builtin_amdgcn_wmma_f32_16x16x4_f32(
            /*neg_a=*/false, a, /*neg_b=*/false, b,
            /*c_mod=*/(short)0, acc, /*reuse_a=*/false, /*reuse_b=*/false);
    }

    if (bok) {
        const float bb = bias ? bias[bcol] : 0.0f;
#pragma unroll
        for (int v = 0; v < 8; ++v) {
            const int row = m0 + v + 8 * half;
            C[(long)row * N + bcol] = acc[v] + bb;
        }
    }
}

// ---------------- aggregation: agg = D^-1/2 (A+I) D^-1/2 @ hw -------------

// self-loop term: agg[i][c] = hw[i][c] * dis[i]^2   (C == 64 fixed)
__global__ void k_self_init(const float* __restrict__ hw, const float* __restrict__ dis,
                            float* __restrict__ agg, int n) {
    long t = (long)blockIdx.x * blockDim.x + threadIdx.x;
    if (t < (long)n * HIDC) {
        int i = (int)(t >> 6);
        float di = dis[i];
        agg[t] = hw[t] * di * di;
    }
}

// edges: 16 threads/edge, float4 per thread, native global_atomic_add_f32
__global__ void k_scatter(const float* __restrict__ hw, const float* __restrict__ dis,
                          const int* __restrict__ src, const int* __restrict__ dst,
                          float* __restrict__ agg, int E) {
    long t = (long)blockIdx.x * blockDim.x + threadIdx.x;
    int e = (int)(t >> 4);
    if (e >= E) return;
    int g = (int)(t & 15);
    int s = src[e];
    int d = dst[e];
    float w = dis[s] * dis[d];
    const float4 v = *((const float4*)(hw + (long)s * HIDC) + g);
    float* ap = agg + (long)d * HIDC + g * 4;
    atomicAdd(ap + 0, v.x * w);
    atomicAdd(ap + 1, v.y * w);
    atomicAdd(ap + 2, v.z * w);
    atomicAdd(ap + 3, v.w * w);
}

__global__ void k_bias_relu(float* __restrict__ h, const float* __restrict__ bias, int n) {
    long t = (long)blockIdx.x * blockDim.x + threadIdx.x;
    if (t < (long)n * HIDC) {
        int c = (int)(t & (HIDC - 1));
        float v = h[t] + bias[c];
        h[t] = v > 0.f ? v : 0.f;
    }
}

// ---------------- driver ----------------

extern "C" void kernel_launch(void* const* d_in, const int* in_sizes, int n_in,
                              void* d_out, int out_size, void* d_ws, size_t ws_size,
                              hipStream_t stream) {
    const float* x  = (const float*)d_in[0];
    const int*   ei = (const int*)d_in[1];
    const float* W1 = (const float*)d_in[2];
    const float* b1 = (const float*)d_in[3];
    const float* W2 = (const float*)d_in[4];
    const float* b2 = (const float*)d_in[5];
    const float* Wc = (const float*)d_in[6];
    const float* bc = (const float*)d_in[7];
    float* out = (float*)d_out;

    const int n = N_NODES;
    const int E = in_sizes[1] / 2;
    const int* src = ei;
    const int* dst = ei + E;

    // workspace layout
    char* ws = (char*)d_ws;
    float* dis = (float*)ws;
    size_t off = (((size_t)n * 4) + 255) & ~(size_t)255;
    float* bufA = (float*)(ws + off);                 // hw buffer (n x 64)
    off += (((size_t)n * HIDC * 4) + 255) & ~(size_t)255;
    float* bufB = (float*)(ws + off);                 // agg/h buffer (n x 64)

    const int T = 256;
    const int gN   = (n + T - 1) / T;
    const int gE   = (E + T - 1) / T;
    const int gNC  = (n * HIDC + T - 1) / T;
    const int gE16 = (int)(((long)E * 16 + T - 1) / T);
    const int rowTiles = n / 16;                      // 50000 = 3125 * 16

    // normalization dis = rsqrt(deg)
    k_init_deg<<<gN, T, 0, stream>>>(dis, n);
    k_count_deg<<<gE, T, 0, stream>>>(dst, dis, E);
    k_rsqrt_inplace<<<gN, T, 0, stream>>>(dis, n);

    // ---- layer 1: h1 = relu(agg(x @ W1) + b1) ----
    k_wmma_gemm<<<rowTiles, 128, 0, stream>>>(x, W1, nullptr, bufA, 128, HIDC);
    k_self_init<<<gNC, T, 0, stream>>>(bufA, dis, bufB, n);
    k_scatter<<<gE16, T, 0, stream>>>(bufA, dis, src, dst, bufB, E);
    k_bias_relu<<<gNC, T, 0, stream>>>(bufB, b1, n);

    // ---- layer 2: h2 = relu(agg(h1 @ W2) + b2) ----
    k_wmma_gemm<<<rowTiles, 128, 0, stream>>>(bufB, W2, nullptr, bufA, HIDC, HIDC);
    k_self_init<<<gNC, T, 0, stream>>>(bufA, dis, bufB, n);
    k_scatter<<<gE16, T, 0, stream>>>(bufA, dis, src, dst, bufB, E);
    k_bias_relu<<<gNC, T, 0, stream>>>(bufB, b2, n);

    // ---- classifier: out = h2 @ Wc + bc  (N=10, one col tile, fused bias) ----
    k_wmma_gemm<<<rowTiles, 32, 0, stream>>>(bufB, Wc, bc, out, HIDC, 10);
}